// ChebLayer_30030411333842
// MI455X (gfx1250) — compile-verified
//
#include <hip/hip_runtime.h>

// ---- problem constants (from reference) ----
#define NB   8
#define MV   32768           // mesh vertices
#define FIN  32
#define RANK 5
#define FILT 32
#define EDG  262144
#define ROWS (NB * MV)       // 262144 GEMM rows
#define VEC  (FIN * NB)      // 256 floats per vertex per Chebyshev term
#define TERM_ELEMS ((long)MV * VEC)   // 8,388,608 floats = 33.5 MB

typedef __attribute__((ext_vector_type(2))) float v2f;
typedef __attribute__((ext_vector_type(8))) float v8f;

// GCC-style vector type matching the async-to-LDS builtin's parameter type
// (diagnostic showed: 'int __attribute__((__vector_size__(4*sizeof(int)))) __device__ *')
typedef int v4i_gcc __attribute__((vector_size(16)));
typedef __attribute__((address_space(1))) v4i_gcc gas_v4i;   // global (AS1)
typedef __attribute__((address_space(3))) v4i_gcc las_v4i;   // LDS (AS3)

// ===========================================================================
// CSR construction from COO (deterministic: per-row edge lists sorted by id)
// ===========================================================================
__global__ __launch_bounds__(256) void csr_count(const int* __restrict__ rows,
                                                 int* __restrict__ cnt) {
  int e = blockIdx.x * 256 + threadIdx.x;
  atomicAdd(&cnt[rows[e]], 1);
}

// Single-block exclusive scan of cnt[MV] -> rowptr[MV+1]; cursor gets a copy.
__global__ __launch_bounds__(1024) void csr_scan(const int* __restrict__ cnt,
                                                 int* __restrict__ rowptr,
                                                 int* __restrict__ cursor) {
  __shared__ int part[1024];
  int t = threadIdx.x;
  int base = t * 32;
  int loc[32];
  int s = 0;
#pragma unroll
  for (int i = 0; i < 32; ++i) { loc[i] = s; s += cnt[base + i]; }
  part[t] = s;
  __syncthreads();
  // Hillis-Steele inclusive scan over 1024 per-thread sums
  for (int off = 1; off < 1024; off <<= 1) {
    int v = (t >= off) ? part[t - off] : 0;
    __syncthreads();
    part[t] += v;
    __syncthreads();
  }
  int prefix = (t == 0) ? 0 : part[t - 1];
#pragma unroll
  for (int i = 0; i < 32; ++i) {
    int v = prefix + loc[i];
    rowptr[base + i] = v;
    cursor[base + i] = v;
  }
  if (t == 1023) rowptr[MV] = part[1023];
}

__global__ __launch_bounds__(256) void csr_fill(const int* __restrict__ rows,
                                                int* __restrict__ cursor,
                                                int* __restrict__ eidx) {
  int e = blockIdx.x * 256 + threadIdx.x;
  int p = atomicAdd(&cursor[rows[e]], 1);
  eidx[p] = e;
}

// Insertion-sort each row's (short, ~8-entry) edge list so the float
// accumulation order is deterministic across calls.
__global__ __launch_bounds__(256) void csr_sort_rows(const int* __restrict__ rowptr,
                                                     int* __restrict__ eidx) {
  int r = blockIdx.x * 256 + threadIdx.x;
  int b = rowptr[r], e2 = rowptr[r + 1];
  for (int i = b + 1; i < e2; ++i) {
    int key = eidx[i];
    int j = i - 1;
    while (j >= b && eidx[j] > key) { eidx[j + 1] = eidx[j]; --j; }
    eidx[j + 1] = key;
  }
}

// ===========================================================================
// x [NB, M, FIN] -> X0 [M, FIN, NB]   (T_0)
// ===========================================================================
__global__ __launch_bounds__(256) void cheb_transpose(const float* __restrict__ x,
                                                      float* __restrict__ x0) {
  long t = (long)blockIdx.x * 256 + threadIdx.x;     // t = m*FIN + f
  float tmp[NB];
#pragma unroll
  for (int n = 0; n < NB; ++n) tmp[n] = x[(long)n * (MV * FIN) + t];
  float4* dst = (float4*)(x0 + t * NB);
  dst[0] = make_float4(tmp[0], tmp[1], tmp[2], tmp[3]);
  dst[1] = make_float4(tmp[4], tmp[5], tmp[6], tmp[7]);
}

__global__ __launch_bounds__(256) void cheb_bias_init(float* __restrict__ out,
                                                      const float* __restrict__ bias) {
  long t = (long)blockIdx.x * 256 + threadIdx.x;
  out[t] = bias[t & (FILT - 1)];
}

// ===========================================================================
// Pull-style SpMM fused with the Chebyshev recurrence:
//   dst[r,:] = scale * sum_{e in row r} vals[e]*src[cols[e],:]  -  pscale*prev[r,:]
// One wave per vertex row; lane owns 8 floats (two B128 gathers per edge).
// No float atomics; edge order fixed by sorted CSR => deterministic.
// ===========================================================================
__global__ __launch_bounds__(256) void cheb_spmm_gather(
    const int* __restrict__ rowptr, const int* __restrict__ eidx,
    const int* __restrict__ cols, const float* __restrict__ vals,
    const float* __restrict__ src, const float* __restrict__ prev,
    float* __restrict__ dst, float scale, float pscale) {
  int wave = threadIdx.x >> 5;
  int lane = threadIdx.x & 31;
  int r = blockIdx.x * 8 + wave;
  int beg = rowptr[r], end = rowptr[r + 1];
  long off = (long)lane * 8;

  float4 a0 = make_float4(0.f, 0.f, 0.f, 0.f);
  float4 a1 = make_float4(0.f, 0.f, 0.f, 0.f);
  for (int j = beg; j < end; ++j) {
    int e = eidx[j];
    float v = vals[e];
    int c = cols[e];
    const float4* s = (const float4*)(src + (long)c * VEC + off);
    float4 s0 = s[0], s1 = s[1];
    a0.x += v * s0.x; a0.y += v * s0.y; a0.z += v * s0.z; a0.w += v * s0.w;
    a1.x += v * s1.x; a1.y += v * s1.y; a1.z += v * s1.z; a1.w += v * s1.w;
  }

  float4 o0 = make_float4(scale * a0.x, scale * a0.y, scale * a0.z, scale * a0.w);
  float4 o1 = make_float4(scale * a1.x, scale * a1.y, scale * a1.z, scale * a1.w);
  long dbase = (long)r * VEC + off;
  if (pscale != 0.0f) {
    const float4* p = (const float4*)(prev + dbase);
    float4 p0 = p[0], p1 = p[1];
    o0.x -= pscale * p0.x; o0.y -= pscale * p0.y; o0.z -= pscale * p0.z; o0.w -= pscale * p0.w;
    o1.x -= pscale * p1.x; o1.y -= pscale * p1.y; o1.z -= pscale * p1.z; o1.w -= pscale * p1.w;
  }
  ((float4*)(dst + dbase))[0] = o0;
  ((float4*)(dst + dbase))[1] = o1;
}

// ===========================================================================
// Per-rank GEMM accumulate via V_WMMA_F32_16X16X4_F32.
// out[r, c] += sum_f Tk[m, f, n] * kernel[f*RANK + rank, c],  r = n*M + m.
// Weight slice (4 KB) staged to LDS via async-to-LDS B128 when available.
// ===========================================================================
__global__ __launch_bounds__(256) void cheb_gemm_wmma(const float* __restrict__ Tk,
                                                      const float* __restrict__ kern,
                                                      float* __restrict__ out,
                                                      int rank) {
  __shared__ float Bs[FIN][FILT];
  int tid = threadIdx.x;
#if __has_builtin(__builtin_amdgcn_global_load_async_to_lds_b128)
  {
    int f = tid >> 3, sub = tid & 7;                 // 256 lanes x 16B = 4 KB
    const float* gp = kern + (f * RANK + rank) * FILT + sub * 4;
    __builtin_amdgcn_global_load_async_to_lds_b128(
        (gas_v4i*)gp, (las_v4i*)&Bs[f][sub * 4], 0, 0);
  }
#if __has_builtin(__builtin_amdgcn_s_wait_asynccnt)
  __builtin_amdgcn_s_wait_asynccnt(0);
#else
  asm volatile("s_wait_asynccnt 0x0" ::: "memory");
#endif
#else
  for (int idx = tid; idx < FIN * FILT; idx += 256) {
    int f = idx >> 5, c = idx & 31;
    Bs[f][c] = kern[(f * RANK + rank) * FILT + c];
  }
#endif
  __syncthreads();

  int wave = tid >> 5;
  int lane = tid & 31;
  long tile  = (long)blockIdx.x * 8 + wave;          // 16-row tile id
  long rbase = tile * 16;
  int  n     = (int)(rbase >> 15);                   // rbase / M  (M = 2^15)
  int  mbase = (int)(rbase & (MV - 1));

  // A fragment (f32 16x4): lanes 0-15 -> K{0,1}, lanes 16-31 -> K{2,3}
  int mrow = lane & 15;
  int koff = (lane >> 4) << 1;
  const float* arow = Tk + (long)(mbase + mrow) * VEC + n;   // element f at arow[f*NB]

  // C fragment: VGPR i -> row i (+8 for lanes 16-31), col = lane&15
  int crow = (lane >> 4) << 3;
  int ccol = lane & 15;

  v8f c0, c1;
#pragma unroll
  for (int i = 0; i < 8; ++i) {
    long r = rbase + i + crow;
    c0[i] = out[r * FILT + ccol];
    c1[i] = out[r * FILT + ccol + 16];
  }

#pragma unroll
  for (int ks = 0; ks < FIN / 4; ++ks) {
    int k0 = ks * 4;
    v2f a;
    a.x = arow[(long)(k0 + koff) * NB];
    a.y = arow[(long)(k0 + koff + 1) * NB];
    v2f b0, b1;                                      // B 4x16: lanes 0-15 K{0,1}, 16-31 K{2,3}
    b0.x = Bs[k0 + koff][ccol];
    b0.y = Bs[k0 + koff + 1][ccol];
    b1.x = Bs[k0 + koff][ccol + 16];
    b1.y = Bs[k0 + koff + 1][ccol + 16];
    c0 = __builtin_amdgcn_wmma_f32_16x16x4_f32(false, a, false, b0, (short)0, c0, false, false);
    c1 = __builtin_amdgcn_wmma_f32_16x16x4_f32(false, a, false, b1, (short)0, c1, false, false);
  }

#pragma unroll
  for (int i = 0; i < 8; ++i) {
    long r = rbase + i + crow;
    out[r * FILT + ccol]      = c0[i];
    out[r * FILT + ccol + 16] = c1[i];
  }
}

// ===========================================================================
extern "C" void kernel_launch(void* const* d_in, const int* in_sizes, int n_in,
                              void* d_out, int out_size, void* d_ws, size_t ws_size,
                              hipStream_t stream) {
  const float* x    = (const float*)d_in[0];
  const float* vals = (const float*)d_in[1];
  const float* kern = (const float*)d_in[2];
  const float* bias = (const float*)d_in[3];
  const int*   rows = (const int*)d_in[4];
  const int*   cols = (const int*)d_in[5];
  float* out = (float*)d_out;

  float* A = (float*)d_ws;          // rotating Chebyshev term buffers (33.5 MB each)
  float* B = A + TERM_ELEMS;
  float* C = B + TERM_ELEMS;
  int* cnt    = (int*)(C + TERM_ELEMS);
  int* rowptr = cnt + MV;
  int* cursor = rowptr + (MV + 1);
  int* eidx   = cursor + MV;

  const int GEMM_BLOCKS  = ROWS / 16 / 8;            // 2048
  const int SPMM_BLOCKS  = MV / 8;                   // 4096 (1 wave per row)
  const int EDGE_BLOCKS  = EDG / 256;                // 1024
  const int SORT_BLOCKS  = MV / 256;                 // 128
  const int XPOSE_BLOCKS = MV * FIN / 256;           // 4096
  const int BIAS_BLOCKS  = ROWS * FILT / 256;        // 32768

  // ---- deterministic CSR from COO ----
  (void)hipMemsetAsync(cnt, 0, MV * sizeof(int), stream);
  csr_count<<<EDGE_BLOCKS, 256, 0, stream>>>(rows, cnt);
  csr_scan<<<1, 1024, 0, stream>>>(cnt, rowptr, cursor);
  csr_fill<<<EDGE_BLOCKS, 256, 0, stream>>>(rows, cursor, eidx);
  csr_sort_rows<<<SORT_BLOCKS, 256, 0, stream>>>(rowptr, eidx);

  // ---- T0 ----
  cheb_transpose<<<XPOSE_BLOCKS, 256, 0, stream>>>(x, A);
  cheb_bias_init<<<BIAS_BLOCKS, 256, 0, stream>>>(out, bias);
  cheb_gemm_wmma<<<GEMM_BLOCKS, 256, 0, stream>>>(A, kern, out, 0);

  // ---- T1 = L T0 ----
  cheb_spmm_gather<<<SPMM_BLOCKS, 256, 0, stream>>>(rowptr, eidx, cols, vals,
                                                    A, A, B, 1.0f, 0.0f);
  cheb_gemm_wmma<<<GEMM_BLOCKS, 256, 0, stream>>>(B, kern, out, 1);

  // ---- T2 = 2 L T1 - T0 ----
  cheb_spmm_gather<<<SPMM_BLOCKS, 256, 0, stream>>>(rowptr, eidx, cols, vals,
                                                    B, A, C, 2.0f, 1.0f);
  cheb_gemm_wmma<<<GEMM_BLOCKS, 256, 0, stream>>>(C, kern, out, 2);

  // ---- T3 = 2 L T2 - T1 ----
  cheb_spmm_gather<<<SPMM_BLOCKS, 256, 0, stream>>>(rowptr, eidx, cols, vals,
                                                    C, B, A, 2.0f, 1.0f);
  cheb_gemm_wmma<<<GEMM_BLOCKS, 256, 0, stream>>>(A, kern, out, 3);

  // ---- T4 = 2 L T3 - T2 ----
  cheb_spmm_gather<<<SPMM_BLOCKS, 256, 0, stream>>>(rowptr, eidx, cols, vals,
                                                    A, C, B, 2.0f, 1.0f);
  cheb_gemm_wmma<<<GEMM_BLOCKS, 256, 0, stream>>>(B, kern, out, 4);
}